// QWTInverse_57123065037233
// MI455X (gfx1250) — compile-verified
//
#include <hip/hip_runtime.h>

typedef float v2f __attribute__((ext_vector_type(2)));
typedef float v8f __attribute__((ext_vector_type(8)));

// ---------------------------------------------------------------------------
// Bicubic (A=-0.75) 2x-upsample tap weights at |d| = 0.25, 0.75, 1.25, 1.75.
// All exact binary fractions.  Weight of source sample (base+k) for output
// sample n of a 16-wide tile with base = origin/2 - 2 (valid k: 0..11):
//   even n=2t: taps k=t..t+3   -> [w175, w075, w025, w125]
//   odd  n=2t+1: taps k=t+1..t+4 -> [w125, w025, w075, w175]
// ---------------------------------------------------------------------------
constexpr float upw_c(int k, int n) {
    const float W025 = 0.87890625f, W075 = 0.26171875f;
    const float W125 = -0.10546875f, W175 = -0.03515625f;
    int t = n >> 1;
    int i = k - t - (n & 1);
    float we = (i == 0) ? W175 : (i == 1) ? W075 : (i == 2) ? W025 : (i == 3) ? W125 : 0.0f;
    float wo = (i == 0) ? W125 : (i == 1) ? W025 : (i == 2) ? W075 : (i == 3) ? W175 : 0.0f;
    return (n & 1) ? wo : we;
}

// Per-(K-slice, lane) WMMA weight operand table.  For slice s, a lane supplies
// K = 4s + 2*(lane>>4) + {0,1} at row/col index (lane & 15).  The same values
// serve as the stage-A B-operand and the stage-B A-operand.
struct WPair { float x, y; };
struct WTab  { WPair w[3][32]; };
constexpr WTab make_wtab() {
    WTab t{};
    for (int s = 0; s < 3; ++s)
        for (int lane = 0; lane < 32; ++lane) {
            int n = lane & 15;
            int ka = 4 * s + ((lane >> 4) << 1);
            t.w[s][lane].x = upw_c(ka, n);
            t.w[s][lane].y = upw_c(ka + 1, n);
        }
    return t;
}
__constant__ WTab WTAB = make_wtab();

// Monotonic float <-> uint mapping for atomic min/max.
__device__ __forceinline__ unsigned fenc(float f) {
    unsigned u = __float_as_uint(f);
    return (u & 0x80000000u) ? ~u : (u | 0x80000000u);
}
__device__ __forceinline__ float fdec(unsigned k) {
    return (k & 0x80000000u) ? __uint_as_float(k ^ 0x80000000u) : __uint_as_float(~k);
}

// ---------------------------------------------------------------------------
// Kernel 0: filter sums -> 16 combine coefficients; init min/max keys.
// ---------------------------------------------------------------------------
__global__ void k_prep(const float* __restrict__ gl, const float* __restrict__ gh,
                       const float* __restrict__ fl, const float* __restrict__ fh,
                       float* __restrict__ wsf) {
    float a = 0.f, b = 0.f, c = 0.f, d = 0.f;
    for (int i = 0; i < 16; ++i) { a += gl[i]; b += gh[i]; c += fl[i]; d += fh[i]; }
    // chunk order within each source: [·g·g, ·f·g, ·g·f, ·f·f]
    wsf[0]  = a * a;          wsf[1]  = c * a;  wsf[2]  = a * c;  wsf[3]  = c * c;  // LL
    wsf[4]  = a * b + c * b;  wsf[5]  = 0.f;    wsf[6]  = a * d;  wsf[7]  = c * d;  // LH (lfhg dead)
    wsf[8]  = b * a;          wsf[9]  = d * a;  wsf[10] = b * c;  wsf[11] = d * c;  // HL
    wsf[12] = b * b;          wsf[13] = d * b;  wsf[14] = b * d;  wsf[15] = d * d;  // HH
    unsigned* keys = (unsigned*)wsf;
    keys[16] = 0xFFFFFFFFu;   // min key
    keys[17] = 0u;            // max key
}

// ---------------------------------------------------------------------------
// Kernel 1: Z[n,k,r,c] = sum_t coef * {LL,LH,HL,HH}[n, k+8t, r, c]
// Z shape (8,8,256,256); float4 vectorized.
// ---------------------------------------------------------------------------
__global__ __launch_bounds__(256) void k_combine(const float* __restrict__ LL,
                                                 const float* __restrict__ Yh,
                                                 const float* __restrict__ coef,
                                                 float* __restrict__ Z) {
    int tid = blockIdx.x * 256 + threadIdx.x;        // 1,048,576 threads
    int e = tid << 2;                                // element index
    int plane = e >> 16;                             // 0..63  (= n*8 + k)
    int pix = e & 65535;
    int n = plane >> 3, k = plane & 7;
    float c[16];
#pragma unroll
    for (int i = 0; i < 16; ++i) c[i] = coef[i];
    float4 acc = make_float4(0.f, 0.f, 0.f, 0.f);
#pragma unroll
    for (int t = 0; t < 4; ++t) {
        int ch = n * 32 + k + 8 * t;
        float4 v = *(const float4*)(LL + (((size_t)ch) << 16) + pix);
        float w = c[t];
        acc.x = fmaf(w, v.x, acc.x); acc.y = fmaf(w, v.y, acc.y);
        acc.z = fmaf(w, v.z, acc.z); acc.w = fmaf(w, v.w, acc.w);
#pragma unroll
        for (int bnd = 0; bnd < 3; ++bnd) {
            float4 u = *(const float4*)(Yh + (((size_t)(ch * 3 + bnd)) << 16) + pix);
            float wc = c[(bnd + 1) * 4 + t];
            acc.x = fmaf(wc, u.x, acc.x); acc.y = fmaf(wc, u.y, acc.y);
            acc.z = fmaf(wc, u.z, acc.z); acc.w = fmaf(wc, u.w, acc.w);
        }
    }
    *(float4*)(Z + (((size_t)plane) << 16) + pix) = acc;
}

// ---------------------------------------------------------------------------
// Kernel 2: 2x bicubic upsample (8,8,256,256) -> (8,8,512,512) via WMMA.
// One wave per 16x16 output tile; two GEMM stages (K=12 as 3x V_WMMA 16x16x4).
// Fused global min/max (block reduce + atomic on encoded keys).
// ---------------------------------------------------------------------------
__global__ __launch_bounds__(256) void k_upsample(const float* __restrict__ Z,
                                                  float* __restrict__ Y,
                                                  unsigned* __restrict__ keys) {
    __shared__ float zbuf[8][16 * 17];
    __shared__ float tbuf[8][16 * 17];
    __shared__ float rminS[256];
    __shared__ float rmaxS[256];

    const int lane = threadIdx.x & 31;
    const int wv   = threadIdx.x >> 5;
    const int tile = blockIdx.x * 8 + wv;            // 65536 tiles total
    const int plane = tile >> 10;                    // 1024 tiles per image
    const int t2 = tile & 1023;
    const int ti = t2 >> 5, tj = t2 & 31;
    const int i0 = ti << 4, j0 = tj << 4;
    const int p0 = (i0 >> 1) - 2;                    // source row base
    const int q0 = (j0 >> 1) - 2;                    // source col base

    const float* zp = Z + (((size_t)plane) << 16);
    float* yp = Y + (((size_t)plane) << 18);

    float* zb = zbuf[wv];
    float* tb = tbuf[wv];

    // Cooperative load of clamped 16x16 Z footprint (only cols 0..11 used).
#pragma unroll
    for (int it = 0; it < 8; ++it) {
        int e = it * 32 + lane;
        int m = e >> 4, kk = e & 15;
        int zr = min(max(p0 + m, 0), 255);
        int zc = min(max(q0 + kk, 0), 255);
        zb[m * 17 + kk] = zp[zr * 256 + zc];
    }

    // Per-lane weight operands for the three K-slices (coalesced 8B loads).
    v2f w0, w1, w2;
    w0.x = WTAB.w[0][lane].x;  w0.y = WTAB.w[0][lane].y;
    w1.x = WTAB.w[1][lane].x;  w1.y = WTAB.w[1][lane].y;
    w2.x = WTAB.w[2][lane].x;  w2.y = WTAB.w[2][lane].y;

    __syncthreads();

    const int mrow = lane & 15;    // A row (M) / B col (N) for this lane
    const int hi = lane >> 4;      // half-wave selector (adds +2 to K, +8 to M)

    // Stage A: T[m, n] = sum_k Z[p0+m, q0+k] * Wcol[k, n]    (column upsample)
    v8f t16 = {0.f, 0.f, 0.f, 0.f, 0.f, 0.f, 0.f, 0.f};
    {
        v2f A;
        A.x = zb[mrow * 17 + (hi << 1) + 0];  A.y = zb[mrow * 17 + (hi << 1) + 1];
        t16 = __builtin_amdgcn_wmma_f32_16x16x4_f32(false, A, false, w0, (short)0, t16, false, false);
        A.x = zb[mrow * 17 + (hi << 1) + 4];  A.y = zb[mrow * 17 + (hi << 1) + 5];
        t16 = __builtin_amdgcn_wmma_f32_16x16x4_f32(false, A, false, w1, (short)0, t16, false, false);
        A.x = zb[mrow * 17 + (hi << 1) + 8];  A.y = zb[mrow * 17 + (hi << 1) + 9];
        t16 = __builtin_amdgcn_wmma_f32_16x16x4_f32(false, A, false, w2, (short)0, t16, false, false);
    }
    // C-layout store: VGPR r -> row r (+8 for upper half-wave), col = lane&15.
#pragma unroll
    for (int r = 0; r < 8; ++r)
        tb[(r + (hi << 3)) * 17 + mrow] = t16[r];
    __syncthreads();

    // Stage B: y[m, n] = sum_k Wrow[m, k] * T[k, n]          (row upsample)
    v8f y16 = {0.f, 0.f, 0.f, 0.f, 0.f, 0.f, 0.f, 0.f};
    {
        v2f B;
        int kb;
        kb = (hi << 1) + 0;
        B.x = tb[kb * 17 + mrow];  B.y = tb[(kb + 1) * 17 + mrow];
        y16 = __builtin_amdgcn_wmma_f32_16x16x4_f32(false, w0, false, B, (short)0, y16, false, false);
        kb = (hi << 1) + 4;
        B.x = tb[kb * 17 + mrow];  B.y = tb[(kb + 1) * 17 + mrow];
        y16 = __builtin_amdgcn_wmma_f32_16x16x4_f32(false, w1, false, B, (short)0, y16, false, false);
        kb = (hi << 1) + 8;
        B.x = tb[kb * 17 + mrow];  B.y = tb[(kb + 1) * 17 + mrow];
        y16 = __builtin_amdgcn_wmma_f32_16x16x4_f32(false, w2, false, B, (short)0, y16, false, false);
    }

    float mn = 3.402823466e38f, mx = -3.402823466e38f;
#pragma unroll
    for (int r = 0; r < 8; ++r) {
        float v = y16[r];
        yp[(size_t)(i0 + r + (hi << 3)) * 512 + j0 + mrow] = v;
        mn = fminf(mn, v);
        mx = fmaxf(mx, v);
    }

    // Block min/max reduction, then one atomic pair per block.
    rminS[threadIdx.x] = mn;
    rmaxS[threadIdx.x] = mx;
    __syncthreads();
    for (int off = 128; off > 0; off >>= 1) {
        if (threadIdx.x < off) {
            rminS[threadIdx.x] = fminf(rminS[threadIdx.x], rminS[threadIdx.x + off]);
            rmaxS[threadIdx.x] = fmaxf(rmaxS[threadIdx.x], rmaxS[threadIdx.x + off]);
        }
        __syncthreads();
    }
    if (threadIdx.x == 0) {
        atomicMin(&keys[0], fenc(rminS[0]));
        atomicMax(&keys[1], fenc(rmaxS[0]));
    }
}

// ---------------------------------------------------------------------------
// Kernel 3: in-place normalize d_out: (y - mn) / (mx - mn).
// ---------------------------------------------------------------------------
__global__ __launch_bounds__(256) void k_norm(float* __restrict__ out,
                                              const unsigned* __restrict__ keys) {
    size_t i = ((size_t)(blockIdx.x * 256 + threadIdx.x)) << 2;
    float mn = fdec(keys[0]);
    float mx = fdec(keys[1]);
    float inv = 1.0f / (mx - mn);
    float4 v = *(float4*)(out + i);
    v.x = (v.x - mn) * inv;
    v.y = (v.y - mn) * inv;
    v.z = (v.z - mn) * inv;
    v.w = (v.w - mn) * inv;
    *(float4*)(out + i) = v;
}

extern "C" void kernel_launch(void* const* d_in, const int* in_sizes, int n_in,
                              void* d_out, int out_size, void* d_ws, size_t ws_size,
                              hipStream_t stream) {
    const float* LL = (const float*)d_in[0];   // (8,32,256,256)
    const float* Yh = (const float*)d_in[1];   // (8,32,3,256,256)
    const float* gl = (const float*)d_in[2];
    const float* gh = (const float*)d_in[3];
    const float* fl = (const float*)d_in[4];
    const float* fh = (const float*)d_in[5];
    float* out = (float*)d_out;                // (8,8,512,512) = 16,777,216 floats

    float* wsf = (float*)d_ws;
    unsigned* keys = ((unsigned*)d_ws) + 16;
    float* Zbuf = wsf + 64;                    // (8,8,256,256) = 16 MB scratch

    k_prep<<<1, 1, 0, stream>>>(gl, gh, fl, fh, wsf);
    k_combine<<<4096, 256, 0, stream>>>(LL, Yh, wsf, Zbuf);   // 4,194,304 elems /4
    k_upsample<<<8192, 256, 0, stream>>>(Zbuf, out, keys);    // 65536 tiles, 8/block
    k_norm<<<16384, 256, 0, stream>>>(out, keys);             // 16,777,216 /4
}